// ScaledDotProductAttention_386547056961
// MI455X (gfx1250) — compile-verified
//
#include <hip/hip_runtime.h>

// SDPA forward for B=2, H=16, S=2048, D=64 (fp32), returning context + full attn.
// Memory-bound on the 537 MB attn write -> fp32 WMMA (V_WMMA_F32_16X16X4_F32)
// keeps reference numerics at zero cost. One fused kernel per 16-row query tile.
// Streaming attn stores / mask loads use non-temporal hints so the 192 MB L2
// keeps the K/V working set (1 MB per (b,h), reused by 128 workgroups).

typedef __attribute__((ext_vector_type(2))) float v2f;
typedef __attribute__((ext_vector_type(4))) float v4f;
typedef __attribute__((ext_vector_type(8))) float v8f;

#define S_LEN 2048
#define DK 64
#define NEG_INF -1.0e9f

// WMMA f32 16x16x4 fragment layouts (ISA 7.12.2, wave32):
//   A (16x4): lane l holds row m = l&15; VGPR v holds k = 4*t + v + 2*(l>=16)
//   B (4x16): lane l holds col n = l&15; VGPR v holds k = v + 2*(l>=16)
//   C/D (16x16): VGPR r, lanes 0-15 -> (m=r, n=l), lanes 16-31 -> (m=r+8, n=l-16)

__global__ __launch_bounds__(256, 1)
void sdpa_fused_kernel(const float* __restrict__ Q, const float* __restrict__ K,
                       const float* __restrict__ V,
                       const unsigned char* __restrict__ mask,  // jnp.bool_ = 1 byte
                       float* __restrict__ ctx, float* __restrict__ attn) {
    __shared__ float sc[16 * S_LEN];      // 128 KB: score tile -> prob tile
    __shared__ float part[8 * 16 * DK];   //  32 KB: per-wave context partials

    const int bh    = blockIdx.y;          // 0..31  (b*H + h)
    const int qbase = blockIdx.x * 16;     // query-row tile base
    const int tid   = threadIdx.x;
    const int lane  = tid & 31;
    const int wave  = tid >> 5;            // 8 waves
    const int hi    = lane >> 4;           // upper lane half
    const int lr    = lane & 15;

    const size_t qkv_base  = (size_t)bh * S_LEN * DK;
    const size_t mask_base = (size_t)bh * S_LEN * S_LEN;

    // ---------------- Phase A: scores = mask ? -1e9 : (Q K^T)/8 -> LDS -------
    v2f aQ[16];
    {
        const float* qrow = Q + qkv_base + (size_t)(qbase + lr) * DK + 2 * hi;
        #pragma unroll
        for (int t = 0; t < 16; ++t)
            aQ[t] = *(const v2f*)(qrow + 4 * t);   // k = 4t+2hi, 4t+2hi+1
    }
    const int col0 = wave * 256;               // this wave's 256 key columns
    for (int ct = 0; ct < 16; ++ct) {
        const int ncol = col0 + ct * 16;
        const float* krow = K + qkv_base + (size_t)(ncol + lr) * DK + 2 * hi;
        v8f c = {};
        #pragma unroll
        for (int t = 0; t < 16; ++t) {
            v2f b = *(const v2f*)(krow + 4 * t);
            c = __builtin_amdgcn_wmma_f32_16x16x4_f32(false, aQ[t], false, b,
                                                      (short)0, c, false, false);
        }
        #pragma unroll
        for (int r = 0; r < 8; ++r) {
            const int m = r + 8 * hi;
            const unsigned char mk = __builtin_nontemporal_load(
                &mask[mask_base + (size_t)(qbase + m) * S_LEN + ncol + lr]);
            float s = c[r] * 0.125f;           // 1/sqrt(64)
            s = mk ? NEG_INF : s;
            sc[m * S_LEN + ncol + lr] = s;
        }
    }
    __syncthreads();

    // ---------------- Phase B: row softmax; probs -> LDS + global attn -------
    #pragma unroll
    for (int rr = 0; rr < 2; ++rr) {
        const int row = wave * 2 + rr;         // 8 waves x 2 rows = 16 rows
        float* srow = sc + row * S_LEN;
        float m = -3.4e38f;
        for (int i = lane; i < S_LEN; i += 32) m = fmaxf(m, srow[i]);
        #pragma unroll
        for (int off = 16; off > 0; off >>= 1) m = fmaxf(m, __shfl_xor(m, off, 32));
        float l = 0.f;
        for (int i = lane; i < S_LEN; i += 32) {
            float e = __expf(srow[i] - m);
            srow[i] = e;
            l += e;
        }
        #pragma unroll
        for (int off = 16; off > 0; off >>= 1) l += __shfl_xor(l, off, 32);
        const float inv = 1.f / l;
        float* arow = attn + mask_base + (size_t)(qbase + row) * S_LEN;
        for (int i = lane; i < S_LEN; i += 32) {
            const float p = srow[i] * inv;
            srow[i] = p;
            __builtin_nontemporal_store(p, &arow[i]);  // streaming, bypass L2 retain
        }
    }
    __syncthreads();

    // ---------------- Phase C: context partials = P_slice @ V ----------------
    v8f acc[4] = {};                            // 4 d-tiles of 16
    const float* vb = V + qkv_base + lr;
    for (int k0 = col0; k0 < col0 + 256; k0 += 4) {
        v2f a = *(const v2f*)&sc[lr * S_LEN + k0 + 2 * hi];
        const float* v0 = vb + (size_t)(k0 + 2 * hi) * DK;
        #pragma unroll
        for (int dt = 0; dt < 4; ++dt) {
            v2f b = { v0[dt * 16], v0[DK + dt * 16] };   // rows k0+2hi, k0+2hi+1
            acc[dt] = __builtin_amdgcn_wmma_f32_16x16x4_f32(false, a, false, b,
                                                            (short)0, acc[dt],
                                                            false, false);
        }
    }
    #pragma unroll
    for (int dt = 0; dt < 4; ++dt) {
        #pragma unroll
        for (int r = 0; r < 8; ++r)
            part[wave * (16 * DK) + (r + 8 * hi) * DK + dt * 16 + lr] = acc[dt][r];
    }
    __syncthreads();

    // ---------------- Reduce 8 wave partials, write context ------------------
    {
        const int e = tid * 4;                 // 1024 f32 / 256 threads
        v4f sum = *(const v4f*)&part[e];
        #pragma unroll
        for (int w = 1; w < 8; ++w)
            sum += *(const v4f*)&part[w * (16 * DK) + e];
        const int m = e / DK, d = e % DK;      // 4 | 64: no row crossing
        *(v4f*)&ctx[qkv_base + (size_t)(qbase + m) * DK + d] = sum;
    }
}

extern "C" void kernel_launch(void* const* d_in, const int* in_sizes, int n_in,
                              void* d_out, int out_size, void* d_ws, size_t ws_size,
                              hipStream_t stream) {
    (void)in_sizes; (void)n_in; (void)d_ws; (void)ws_size; (void)out_size;
    const float* Q = (const float*)d_in[0];
    const float* K = (const float*)d_in[1];
    const float* V = (const float*)d_in[2];
    const unsigned char* mask = (const unsigned char*)d_in[3];  // jnp.bool_

    float* ctx  = (float*)d_out;                               // [B,H,S,D] first
    float* attn = ctx + (size_t)2 * 16 * S_LEN * DK;           // then [B,H,S,S]

    dim3 grid(S_LEN / 16, 2 * 16);   // (query tiles, B*H)
    dim3 block(256);
    sdpa_fused_kernel<<<grid, block, 0, stream>>>(Q, K, V, mask, ctx, attn);
}